// TTLowRankLinear_3152505995821
// MI455X (gfx1250) — compile-verified
//
#include <hip/hip_runtime.h>

typedef float v2f __attribute__((ext_vector_type(2)));
typedef float v8f __attribute__((ext_vector_type(8)));

// ---------------------------------------------------------------------------
// Problem constants (from the reference)
// ---------------------------------------------------------------------------
#define KDIM 4096          // IN_FEATURES
#define NDIM 4096          // OUT_FEATURES
#define MDIM 16384         // 4 * 4096 rows of x

// GEMM tiling
#define BM 128
#define BN 128
#define BK 32
#define LDA 36             // padded LDS stride (floats) for A tile: 144B rows (16B aligned, 16 rows hit distinct banks)
#define LDB 136            // padded LDS stride (floats) for B tile: 544B rows (16B aligned, k/k+2 rows hit disjoint banks)
#define NITER (KDIM / BK)  // 128 k-tiles

// ---------------------------------------------------------------------------
// Kernel 1: materialize W (4096x4096 fp32) from the three TT cores.
//   row = i0*256 + o0*16 + i1 ; col = o1*256 + i2*16 + o2
//   W[row,col] = sum_{r1,r2} c0[0,i0,o0,r1] * c1[r1,i1,o1,r2] * c2[r2,i2,o2,0]
// ---------------------------------------------------------------------------
__global__ __launch_bounds__(256)
void tt_build_w(const float* __restrict__ c0,
                const float* __restrict__ c1,
                const float* __restrict__ c2,
                float* __restrict__ W) {
    const int idx = blockIdx.x * 256 + threadIdx.x;     // 0 .. 4096*4096-1
    const int row = idx >> 12;
    const int col = idx & 4095;
    const int i0 = row >> 8, o0 = (row >> 4) & 15, i1 = row & 15;
    const int o1 = col >> 8, i2 = (col >> 4) & 15, o2 = col & 15;

    const float* a = c0 + ((i0 << 4) + o0) * 8;         // [r1], r0 == 1

    float t[8];
#pragma unroll
    for (int r2 = 0; r2 < 8; ++r2) t[r2] = 0.0f;
#pragma unroll
    for (int r1 = 0; r1 < 8; ++r1) {
        const float av = a[r1];
        const float* b = c1 + (((r1 << 4) + i1) * 16 + o1) * 8;  // [r2]
#pragma unroll
        for (int r2 = 0; r2 < 8; ++r2) t[r2] = fmaf(av, b[r2], t[r2]);
    }
    float acc = 0.0f;
#pragma unroll
    for (int r2 = 0; r2 < 8; ++r2)
        acc = fmaf(t[r2], c2[((r2 << 4) + i2) * 16 + o2], acc);

    W[idx] = acc;
}

// ---------------------------------------------------------------------------
// Async global->LDS copy, 16B per lane (GVS form: saddr base + 32-bit voffset).
// LDS address comes from the VDST VGPR; tracked by ASYNCcnt, retires in order.
// ---------------------------------------------------------------------------
__device__ __forceinline__ void async_copy_b128(unsigned lds_byte_addr,
                                                const void* sbase,
                                                unsigned global_byte_off) {
    asm volatile("global_load_async_to_lds_b128 %0, %1, %2"
                 :
                 : "v"(lds_byte_addr), "v"(global_byte_off), "s"(sbase)
                 : "memory");
}

// ---------------------------------------------------------------------------
// Kernel 2: out = x @ W via V_WMMA_F32_16X16X4_F32 (full fp32 precision).
// 256 threads = 8 waves; block tile 128x128, BK=32; double-buffered LDS fed
// by async DMA. Wave grid 4(m) x 2(n): each wave owns 32x64 of C
// (2x4 accumulators of 16x16 f32, v8f each) -> 64 WMMAs per k-tile per wave.
//
// Fragment layouts (ISA 7.12.2, wave32):
//   A 16x4 : lane<16 -> M=lane,    VGPR{0,1} = K{0,1};
//            lane>=16-> M=lane-16, VGPR{0,1} = K{2,3}   (contiguous float2)
//   B 4x16 : VGPR0 lanes0-15 = row K0, lanes16-31 = row K2; VGPR1 = K1/K3
//   C/D    : VGPR v, lanes0-15 -> M=v, lanes16-31 -> M=v+8, N=lane%16
// ---------------------------------------------------------------------------
__global__ __launch_bounds__(256)
void tt_gemm_wmma(const float* __restrict__ X,
                  const float* __restrict__ W,
                  float* __restrict__ Out) {
    __shared__ float As[2][BM * LDA];   // 2 x 18KB
    __shared__ float Bs[2][BK * LDB];   // 2 x 17KB

    const int tid  = threadIdx.x;
    const int lane = tid & 31;
    const int wid  = tid >> 5;            // 0..7
    const int wm   = (wid & 3) * 32;      // wave m-offset inside block tile
    const int wn   = (wid >> 2) * 64;     // wave n-offset inside block tile
    const int half = lane >> 4;           // 0 | 1
    const int lrow = lane & 15;

    const int mBlock = blockIdx.y * BM;
    const int nBlock = blockIdx.x * BN;

    // LDS byte offsets (low 32 bits of an LDS generic address == LDS offset)
    const unsigned asBase0 = (unsigned)(uintptr_t)&As[0][0];
    const unsigned asBase1 = (unsigned)(uintptr_t)&As[1][0];
    const unsigned bsBase0 = (unsigned)(uintptr_t)&Bs[0][0];
    const unsigned bsBase1 = (unsigned)(uintptr_t)&Bs[1][0];

    v8f acc[2][4];
#pragma unroll
    for (int i = 0; i < 2; ++i)
#pragma unroll
        for (int j = 0; j < 4; ++j)
            acc[i][j] = (v8f){};

    // per-thread DMA assignments (16B each, 4 per tile for A and for B)
    const int aVec = tid & 7;             // k chunk: aVec*4 floats
    const int aRow = tid >> 3;            // 0..31 (rows r, r+32, r+64, r+96)
    const int bVec = tid & 31;            // n chunk: bVec*4 floats
    const int bRow = tid >> 5;            // 0..7   (rows r, r+8, r+16, r+24)

    // issue one k-tile's async loads into buffer `buf`
    auto issue_tile = [&](int kBase, unsigned asB, unsigned bsB) {
#pragma unroll
        for (int r = 0; r < 4; ++r) {
            const int m = aRow + r * 32;
            const unsigned lds = asB + (unsigned)((m * LDA + aVec * 4) * 4);
            const unsigned go  = (unsigned)(((mBlock + m) * KDIM + kBase + aVec * 4) * 4);
            async_copy_b128(lds, X, go);
        }
#pragma unroll
        for (int r = 0; r < 4; ++r) {
            const int k = bRow + r * 8;
            const unsigned lds = bsB + (unsigned)((k * LDB + bVec * 4) * 4);
            const unsigned go  = (unsigned)(((kBase + k) * NDIM + nBlock + bVec * 4) * 4);
            async_copy_b128(lds, W, go);
        }
    };

    // prologue: tile 0 -> buffer 0
    issue_tile(0, asBase0, bsBase0);

    for (int it = 0; it < NITER; ++it) {
        const int cur = it & 1;
        if (it + 1 < NITER) {
            // prefetch next tile into the other buffer (overlaps with wait+compute)
            issue_tile((it + 1) * BK,
                       cur ? asBase0 : asBase1,
                       cur ? bsBase0 : bsBase1);
            // 16 outstanding, oldest 8 = current tile (async loads retire in order)
            asm volatile("s_wait_asynccnt 8" ::: "memory");
        } else {
            asm volatile("s_wait_asynccnt 0" ::: "memory");
        }
        __syncthreads();                  // all waves' DMA for tile `it` landed

        const float* as = As[cur];
        const float* bs = Bs[cur];

#pragma unroll
        for (int k0 = 0; k0 < BK; k0 += 4) {
            v2f aFrag[2];
#pragma unroll
            for (int i = 0; i < 2; ++i)
                aFrag[i] = *reinterpret_cast<const v2f*>(
                    &as[(wm + i * 16 + lrow) * LDA + k0 + half * 2]);

            v2f bFrag[4];
#pragma unroll
            for (int j = 0; j < 4; ++j) {
                const int n = wn + j * 16 + lrow;
                v2f b;
                b.x = bs[(k0 + half * 2)     * LDB + n];
                b.y = bs[(k0 + half * 2 + 1) * LDB + n];
                bFrag[j] = b;
            }

#pragma unroll
            for (int i = 0; i < 2; ++i)
#pragma unroll
                for (int j = 0; j < 4; ++j)
                    acc[i][j] = __builtin_amdgcn_wmma_f32_16x16x4_f32(
                        /*neg_a=*/false, aFrag[i],
                        /*neg_b=*/false, bFrag[j],
                        /*c_mod=*/(short)0, acc[i][j],
                        /*reuse_a=*/false, /*reuse_b=*/false);
        }
        __syncthreads();                  // buffer safe to overwrite next iter
    }

    // Write back C tiles (lanes 0-15 cover 64B contiguous per store -> coalesced)
#pragma unroll
    for (int i = 0; i < 2; ++i) {
#pragma unroll
        for (int j = 0; j < 4; ++j) {
            const int col = nBlock + wn + j * 16 + lrow;
#pragma unroll
            for (int v = 0; v < 8; ++v) {
                const int rowg = mBlock + wm + i * 16 + v + half * 8;
                Out[(size_t)rowg * NDIM + col] = acc[i][j][v];
            }
        }
    }
}

// ---------------------------------------------------------------------------
// Launch: inputs in setup_inputs() order: x, core0, core1, core2 (all fp32).
// d_ws holds the materialized W (4096*4096*4B = 64MB).
// ---------------------------------------------------------------------------
extern "C" void kernel_launch(void* const* d_in, const int* in_sizes, int n_in,
                              void* d_out, int out_size, void* d_ws, size_t ws_size,
                              hipStream_t stream) {
    const float* x  = (const float*)d_in[0];
    const float* c0 = (const float*)d_in[1];
    const float* c1 = (const float*)d_in[2];
    const float* c2 = (const float*)d_in[3];
    float* out = (float*)d_out;
    float* Wm  = (float*)d_ws;           // 64 MB scratch

    // Build W: 4096*4096 elements, 256 threads/block.
    tt_build_w<<<(KDIM * NDIM) / 256, 256, 0, stream>>>(c0, c1, c2, Wm);

    // GEMM: out(16384x4096) = x(16384x4096) @ W(4096x4096)
    dim3 grid(NDIM / BN, MDIM / BM);     // (32, 128)
    tt_gemm_wmma<<<grid, 256, 0, stream>>>(x, Wm, out);
}